// NodeAttention_58188216926495
// MI455X (gfx1250) — compile-verified
//
#include <hip/hip_runtime.h>
#include <hip/hip_bf16.h>
#include <math.h>

typedef __attribute__((ext_vector_type(2))) float v2f;
typedef __attribute__((ext_vector_type(8))) float v8f;

#define LEAKY_ALPHA 0.2f

// ---------- helpers: monotone float<->int mapping for atomic segment-max ----------
__device__ __forceinline__ int floatToOrderedInt(float f) {
    int i = __float_as_int(f);
    return (i >= 0) ? i : (i ^ 0x7FFFFFFF);
}
__device__ __forceinline__ float orderedIntToFloat(int i) {
    return __int_as_float((i >= 0) ? i : (i ^ 0x7FFFFFFF));
}
// orderedInt(-inf) = 0xFF800000 ^ 0x7FFFFFFF = 0x807FFFFF
#define ORDERED_NEG_INF ((int)0x807FFFFF)

__device__ __forceinline__ float leaky(float v) {
    return v > 0.0f ? v : LEAKY_ALPHA * v;
}

// ---------- 0: init vals (=d_out), denom, segment-max keys ----------
__global__ __launch_bounds__(256) void gat_init(float* __restrict__ vals,
                                                float* __restrict__ denom,
                                                int* __restrict__ m_ord,
                                                int N) {
    int i = blockIdx.x * 256 + threadIdx.x;
    long long tot = (long long)N * 64;
    if (i < tot) vals[i] = 0.0f;
    if (i < N) { denom[i] = 0.0f; m_ord[i] = ORDERED_NEG_INF; }
}

// ---------- 1: h = x @ W via V_WMMA_F32_16X16X4_F32 ----------
// Block = 256 threads = 8 waves; each wave owns a 16-row x 64-col tile of h.
// W staged in LDS pair-major: sWp[k/2][n] = {W[k][n], W[k+1][n]} so each
// B fragment is a single aligned ds_load_b64 into an even VGPR pair.
__global__ __launch_bounds__(256) void gat_gemm(const float* __restrict__ x,
                                                const float* __restrict__ W,
                                                float* __restrict__ h,
                                                int N) {
    __shared__ v2f sWp[128 * 64];   // 64 KB: pair p covers K=2p,2p+1
    for (int idx = threadIdx.x; idx < 128 * 64; idx += 256) {
        int p = idx >> 6, n = idx & 63;
        v2f t;
        t.x = W[(2 * p)     * 64 + n];
        t.y = W[(2 * p + 1) * 64 + n];
        sWp[p * 64 + n] = t;
    }
    __syncthreads();

    const int wave = threadIdx.x >> 5;
    const int lane = threadIdx.x & 31;
    const int half = lane >> 4;      // lanes 0-15 vs 16-31
    const int l16  = lane & 15;

    const int rowBase = blockIdx.x * 128 + wave * 16;
    const int arow    = rowBase + l16;               // A-matrix row for this lane
    // Out-of-range lanes read a clamped valid row; their C rows are never stored.
    const float* __restrict__ xrow =
        x + (size_t)(arow < N ? arow : (N - 1)) * 256;

    v8f c0 = {}, c1 = {}, c2 = {}, c3 = {};

    for (int k0 = 0; k0 < 256; k0 += 4) {
        const int ka = k0 + half * 2;                // A layout: half0 -> K=k0,k0+1 ; half1 -> K=k0+2,k0+3
        v2f a = *(const v2f*)(xrow + ka);            // one global b64 load

        const v2f* wp = sWp + (ka >> 1) * 64;        // pair row for (K=ka, K=ka+1)
        v2f b0 = wp[l16     ];
        v2f b1 = wp[l16 + 16];
        v2f b2 = wp[l16 + 32];
        v2f b3 = wp[l16 + 48];

        c0 = __builtin_amdgcn_wmma_f32_16x16x4_f32(false, a, false, b0, (short)0, c0, false, false);
        c1 = __builtin_amdgcn_wmma_f32_16x16x4_f32(false, a, false, b1, (short)0, c1, false, false);
        c2 = __builtin_amdgcn_wmma_f32_16x16x4_f32(false, a, false, b2, (short)0, c2, false, false);
        c3 = __builtin_amdgcn_wmma_f32_16x16x4_f32(false, a, false, b3, (short)0, c3, false, false);
    }

    // C/D layout: VGPR r holds (M = r + half*8, N = l16)
    #pragma unroll
    for (int r = 0; r < 8; ++r) {
        int m = rowBase + r + half * 8;
        if (m < N) {
            float* hr = h + (size_t)m * 64;
            hr[l16     ] = c0[r];
            hr[l16 + 16] = c1[r];
            hr[l16 + 32] = c2[r];
            hr[l16 + 48] = c3[r];
        }
    }
}

// ---------- 2: per-node attention scores f1 = h@a1+b1 , f2 = h@a2+b2 ----------
__global__ __launch_bounds__(256) void gat_scores(const float* __restrict__ h,
                                                  const float* __restrict__ a1,
                                                  const float* __restrict__ b1,
                                                  const float* __restrict__ a2,
                                                  const float* __restrict__ b2,
                                                  float* __restrict__ f1,
                                                  float* __restrict__ f2,
                                                  int N) {
    int n = blockIdx.x * 256 + threadIdx.x;
    if (n >= N) return;
    const float4* hr = (const float4*)(h + (size_t)n * 64);
    const float4* A1 = (const float4*)a1;
    const float4* A2 = (const float4*)a2;
    float s1 = 0.0f, s2 = 0.0f;
    #pragma unroll
    for (int i = 0; i < 16; ++i) {
        float4 hv = hr[i], x1 = A1[i], x2 = A2[i];
        s1 += hv.x * x1.x + hv.y * x1.y + hv.z * x1.z + hv.w * x1.w;
        s2 += hv.x * x2.x + hv.y * x2.y + hv.z * x2.z + hv.w * x2.w;
    }
    f1[n] = s1 + b1[0];
    f2[n] = s2 + b2[0];
}

// ---------- 3: segment max of edge logits (grouped by src) ----------
__global__ __launch_bounds__(256) void gat_edge_max(const int* __restrict__ esrc,
                                                    const int* __restrict__ edst,
                                                    const float* __restrict__ f1,
                                                    const float* __restrict__ f2,
                                                    int* __restrict__ m_ord,
                                                    int E) {
    int e = blockIdx.x * 256 + threadIdx.x;
    if (e >= E) return;
    int s = esrc[e], d = edst[e];
    float v = leaky(f1[s] + f2[d]);
    atomicMax(&m_ord[s], floatToOrderedInt(v));
}

// ---------- 4: ex = exp(e - m[src]); denom[src] += ex ----------
__global__ __launch_bounds__(256) void gat_edge_exp(const int* __restrict__ esrc,
                                                    const int* __restrict__ edst,
                                                    const float* __restrict__ f1,
                                                    const float* __restrict__ f2,
                                                    const int* __restrict__ m_ord,
                                                    float* __restrict__ ex,
                                                    float* __restrict__ denom,
                                                    int E) {
    int e = blockIdx.x * 256 + threadIdx.x;
    if (e >= E) return;
    int s = esrc[e], d = edst[e];
    float v = leaky(f1[s] + f2[d]);
    float m = orderedIntToFloat(m_ord[s]);   // finite: s has >=1 outgoing edge
    float t = expf(v - m);
    ex[e] = t;
    atomicAdd(&denom[s], t);
}

// ---------- 5: SpMM  vals[src] += (ex/denom[src]) * h[dst]  (64 threads/edge) ----------
__global__ __launch_bounds__(256) void gat_edge_spmm(const int* __restrict__ esrc,
                                                     const int* __restrict__ edst,
                                                     const float* __restrict__ ex,
                                                     const float* __restrict__ denom,
                                                     const float* __restrict__ h,
                                                     float* __restrict__ vals,
                                                     int E) {
    unsigned t = blockIdx.x * 256u + threadIdx.x;
    unsigned e = t >> 6;
    unsigned col = t & 63u;
    if (e >= (unsigned)E) return;
    int s = esrc[e], d = edst[e];
    float coef = ex[e] / denom[s];
    atomicAdd(&vals[(size_t)s * 64 + col], coef * h[(size_t)d * 64 + col]);
}

// ---------- 6: out = elu(vals + bias) ----------
__global__ __launch_bounds__(256) void gat_finish(float* __restrict__ out,
                                                  const float* __restrict__ bias,
                                                  int N) {
    int i = blockIdx.x * 256 + threadIdx.x;
    long long tot = (long long)N * 64;
    if (i >= tot) return;
    float v = out[i] + bias[i & 63];
    out[i] = v > 0.0f ? v : expm1f(v);
}

extern "C" void kernel_launch(void* const* d_in, const int* in_sizes, int n_in,
                              void* d_out, int out_size, void* d_ws, size_t ws_size,
                              hipStream_t stream) {
    const float* x        = (const float*)d_in[0];
    const float* W        = (const float*)d_in[1];
    const float* a1       = (const float*)d_in[2];
    const float* b1       = (const float*)d_in[3];
    const float* a2       = (const float*)d_in[4];
    const float* b2       = (const float*)d_in[5];
    const float* out_bias = (const float*)d_in[6];
    const int*   esrc     = (const int*)d_in[7];
    const int*   edst     = (const int*)d_in[8];

    const int N = in_sizes[0] / 256;   // x is [1, N, 256]
    const int E = in_sizes[7];

    float* out = (float*)d_out;        // doubles as `vals` accumulator

    // workspace carve-up
    char* ws = (char*)d_ws;
    float* h     = (float*)ws;  ws += (size_t)N * 64 * sizeof(float);
    float* f1    = (float*)ws;  ws += (size_t)N * sizeof(float);
    float* f2    = (float*)ws;  ws += (size_t)N * sizeof(float);
    int*   m_ord = (int*)ws;    ws += (size_t)N * sizeof(int);
    float* denom = (float*)ws;  ws += (size_t)N * sizeof(float);
    float* ex    = (float*)ws;  // E floats

    const unsigned nodeVecBlocks = (unsigned)(((long long)N * 64 + 255) / 256);
    const unsigned nodeBlocks    = (unsigned)((N + 255) / 256);
    const unsigned edgeBlocks    = (unsigned)((E + 255) / 256);
    const unsigned spmmBlocks    = (unsigned)(((long long)E * 64 + 255) / 256);
    const unsigned gemmBlocks    = (unsigned)((N + 127) / 128);

    gat_init     <<<nodeVecBlocks, 256, 0, stream>>>(out, denom, m_ord, N);
    gat_gemm     <<<gemmBlocks,    256, 0, stream>>>(x, W, h, N);
    gat_scores   <<<nodeBlocks,    256, 0, stream>>>(h, a1, b1, a2, b2, f1, f2, N);
    gat_edge_max <<<edgeBlocks,    256, 0, stream>>>(esrc, edst, f1, f2, m_ord, E);
    gat_edge_exp <<<edgeBlocks,    256, 0, stream>>>(esrc, edst, f1, f2, m_ord, ex, denom, E);
    gat_edge_spmm<<<spmmBlocks,    256, 0, stream>>>(esrc, edst, ex, denom, h, out, E);
    gat_finish   <<<nodeVecBlocks, 256, 0, stream>>>(out, out_bias, N);
}